// SimpleLocalEncoder_85117661872167
// MI455X (gfx1250) — compile-verified
//
#include <hip/hip_runtime.h>
#include <math.h>

#define NTOK 8
#define D 64
#define DFF 128
#define DM 128
#define EPSF 1.19209290e-7f
#define LDX 68
#define LDQ 196
#define NP 4  // patches per block -> M = 32 rows (two 16-row WMMA tiles)

typedef __attribute__((ext_vector_type(2))) float v2f;
typedef __attribute__((ext_vector_type(8))) float v8f;

// Two 16x16 fp32 output tiles (rows 0-15 and 16-31) sharing B fragments.
// Per-ISA layouts for V_WMMA_F32_16X16X4_F32:
//   A: lane m=lane&15 holds row m; VGPR0/1 = K = half*2, half*2+1 (half=lane>>4)
//   B: VGPR0/1 hold rows K=half*2, half*2+1 at col n0+(lane&15)
//   C/D: VGPR i -> row i + 8*half, col n0+(lane&15)
__device__ inline void wmma_tile2_f32(const float* __restrict__ sA, int lda,
                                      const float* __restrict__ W, int ldw,
                                      int n0, int K, int lane, v8f& c0,
                                      v8f& c1) {
  const int m = lane & 15;
  const int half = lane >> 4;
  c0 = {};
  c1 = {};
  for (int k0 = 0; k0 < K; k0 += 4) {
    const int ka = k0 + half * 2;
    v2f b, a0, a1;
    b.x = W[ka * ldw + n0 + m];
    b.y = W[(ka + 1) * ldw + n0 + m];
    a0.x = sA[m * lda + ka];
    a0.y = sA[m * lda + ka + 1];
    a1.x = sA[(m + 16) * lda + ka];
    a1.y = sA[(m + 16) * lda + ka + 1];
    c0 = __builtin_amdgcn_wmma_f32_16x16x4_f32(false, a0, false, b, (short)0,
                                               c0, false, false);
    c1 = __builtin_amdgcn_wmma_f32_16x16x4_f32(false, a1, false, b, (short)0,
                                               c1, false, false);
  }
}

__global__ __launch_bounds__(128) void local_encoder_kernel(
    const int* __restrict__ tokens, const int* __restrict__ lengths,
    const float* __restrict__ emb, const float* __restrict__ qkv_w,
    const float* __restrict__ w_o, const float* __restrict__ norm1_w,
    const float* __restrict__ norm2_w, const float* __restrict__ ffn_w1,
    const float* __restrict__ ffn_w2, const float* __restrict__ norm_w,
    const float* __restrict__ proj_w, const float* __restrict__ proj_b,
    float* __restrict__ out, int npatch) {
  __shared__ float sX[32 * LDX];    // residual stream x  [32 rows][64]
  __shared__ float sXN[32 * LDX];   // normed / attn-out scratch
  __shared__ float sQKV[32 * LDQ];  // qkv [32][192] / silu(ffn) [32][128]
  __shared__ float sS[512];         // attn probs [4 patch][2 head][8][8]
  __shared__ float sRed[256];
  __shared__ float sRinv[32];
  __shared__ float sPool[NP * 64];

  const int t = threadIdx.x;
  const int lane = t & 31;
  const int wave = t >> 5;
  const int pbase = blockIdx.x * NP;
  const int r16 = t >> 3;  // 0..15 (row within a 16-row tile)
  const int seg = t & 7;   // 8-float segment within a 64-wide row
  const int m = lane & 15;
  const int half = lane >> 4;

  // ---- Phase 0: embedding gather -> sX (32 rows) ----
#pragma unroll
  for (int rr = 0; rr < 2; ++rr) {
    const int row = rr * 16 + r16;
    int patch = pbase + (row >> 3);
    if (patch >= npatch) patch = npatch - 1;  // clamp (keeps EXEC full)
    const int pos = row & 7;
    const int tok = tokens[patch * NTOK + pos];
    const float* e = emb + (size_t)tok * D + seg * 8;
    float* dst = sX + row * LDX + seg * 8;
#pragma unroll
    for (int j = 0; j < 8; ++j) dst[j] = e[j];
  }
  __syncthreads();

  // ---- RMSNorm: sXN = rmsnorm(sX) * w ----
  auto rmsnorm = [&](const float* __restrict__ w) {
#pragma unroll
    for (int rr = 0; rr < 2; ++rr) {
      const int row = rr * 16 + r16;
      const float* xr = sX + row * LDX + seg * 8;
      float ps = 0.f;
#pragma unroll
      for (int j = 0; j < 8; ++j) ps += xr[j] * xr[j];
      sRed[rr * 128 + t] = ps;
    }
    __syncthreads();
    if (t < 32) {
      float s = 0.f;
#pragma unroll
      for (int j = 0; j < 8; ++j) s += sRed[(t >> 4) * 128 + (t & 15) * 8 + j];
      sRinv[(t >> 4) * 16 + (t & 15)] = rsqrtf(s * (1.0f / 64.0f) + EPSF);
    }
    __syncthreads();
#pragma unroll
    for (int rr = 0; rr < 2; ++rr) {
      const int row = rr * 16 + r16;
      const float ri = sRinv[row];
      const float* xr = sX + row * LDX + seg * 8;
      float* dn = sXN + row * LDX + seg * 8;
      const float* ww = w + seg * 8;
#pragma unroll
      for (int j = 0; j < 8; ++j) dn[j] = xr[j] * ri * ww[j];
    }
    __syncthreads();
  };

  rmsnorm(norm1_w);

  // ---- GEMM1: qkv[32x192] = xn @ qkv_w (12 col tiles, 3 per wave) ----
  for (int j = 0; j < 3; ++j) {
    const int n0 = (wave + 4 * j) * 16;
    v8f c0, c1;
    wmma_tile2_f32(sXN, LDX, qkv_w, 192, n0, 64, lane, c0, c1);
#pragma unroll
    for (int i = 0; i < 8; ++i) {
      sQKV[(i + half * 8) * LDQ + n0 + m] = c0[i];
      sQKV[(i + half * 8 + 16) * LDQ + n0 + m] = c1[i];
    }
  }
  __syncthreads();

  // ---- RoPE on q and k (half-dim 16, exponent j/15 per linspace) ----
  if (t < 64) {
    const int pi = t >> 4, h = (t >> 3) & 1, p = t & 7;
    float* rq = sQKV + (pi * 8 + p) * LDQ + h * 32;
    float* rk = rq + 64;
#pragma unroll
    for (int j = 0; j < 16; ++j) {
      const float fr = expf(-9.210340371976184f * (float)j * (1.0f / 15.0f));
      const float th = (float)p * fr;
      const float cs = cosf(th), sn = sinf(th);
      const float q1 = rq[j], q2 = rq[16 + j];
      rq[j] = q1 * cs - q2 * sn;
      rq[16 + j] = q1 * sn + q2 * cs;
      const float k1 = rk[j], k2 = rk[16 + j];
      rk[j] = k1 * cs - k2 * sn;
      rk[16 + j] = k1 * sn + k2 * cs;
    }
  }
  __syncthreads();

  // ---- Attention scores (causal, scaled) ----
  for (int ss = t; ss < 512; ss += 128) {
    const int pi = ss >> 7, r = ss & 127;
    const int h = r >> 6, qp = (r >> 3) & 7, kp = r & 7;
    const float* qr = sQKV + (pi * 8 + qp) * LDQ + h * 32;
    const float* kr = sQKV + (pi * 8 + kp) * LDQ + 64 + h * 32;
    float d = 0.f;
#pragma unroll
    for (int dd = 0; dd < 32; ++dd) d += qr[dd] * kr[dd];
    sS[ss] = (kp <= qp) ? d * 0.17677669529663687f : -1e30f;
  }
  __syncthreads();

  // ---- Softmax over each row of 8 (64 rows) ----
  if (t < 64) {
    float* rp = sS + t * 8;
    float mx = rp[0];
#pragma unroll
    for (int j = 1; j < 8; ++j) mx = fmaxf(mx, rp[j]);
    float sum = 0.f;
#pragma unroll
    for (int j = 0; j < 8; ++j) {
      const float e = expf(rp[j] - mx);
      rp[j] = e;
      sum += e;
    }
    const float inv = 1.0f / sum;
#pragma unroll
    for (int j = 0; j < 8; ++j) rp[j] *= inv;
  }
  __syncthreads();

  // ---- attn @ V -> sXN[32][64] ----
#pragma unroll
  for (int rr = 0; rr < 2; ++rr) {
    const int row = rr * 16 + r16;
    const int pi = row >> 3, qp = row & 7, cb = seg * 8;
#pragma unroll
    for (int cc = 0; cc < 8; ++cc) {
      const int c = cb + cc, h = c >> 5;
      const float* pr = sS + pi * 128 + h * 64 + qp * 8;
      float acc = 0.f;
#pragma unroll
      for (int kp = 0; kp < 8; ++kp)
        acc += pr[kp] * sQKV[(pi * 8 + kp) * LDQ + 128 + c];
      sXN[row * LDX + c] = acc;
    }
  }
  __syncthreads();

  // ---- GEMM3: x += attn_out @ w_o (4 col tiles, 1 per wave) ----
  {
    const int n0 = wave * 16;
    v8f c0, c1;
    wmma_tile2_f32(sXN, LDX, w_o, 64, n0, 64, lane, c0, c1);
#pragma unroll
    for (int i = 0; i < 8; ++i) {
      sX[(i + half * 8) * LDX + n0 + m] += c0[i];
      sX[(i + half * 8 + 16) * LDX + n0 + m] += c1[i];
    }
  }
  __syncthreads();

  rmsnorm(norm2_w);

  // ---- GEMM4: silu(h @ ffn_w1) -> sQKV[32][128] (8 col tiles, 2/wave) ----
  for (int j = 0; j < 2; ++j) {
    const int n0 = (wave + 4 * j) * 16;
    v8f c0, c1;
    wmma_tile2_f32(sXN, LDX, ffn_w1, DFF, n0, 64, lane, c0, c1);
#pragma unroll
    for (int i = 0; i < 8; ++i) {
      const float v0 = c0[i];
      const float v1 = c1[i];
      sQKV[(i + half * 8) * LDQ + n0 + m] = v0 / (1.0f + expf(-v0));
      sQKV[(i + half * 8 + 16) * LDQ + n0 + m] = v1 / (1.0f + expf(-v1));
    }
  }
  __syncthreads();

  // ---- GEMM5: x += silu @ ffn_w2 (K=128, 4 col tiles, 1 per wave) ----
  {
    const int n0 = wave * 16;
    v8f c0, c1;
    wmma_tile2_f32(sQKV, LDQ, ffn_w2, 64, n0, 128, lane, c0, c1);
#pragma unroll
    for (int i = 0; i < 8; ++i) {
      sX[(i + half * 8) * LDX + n0 + m] += c0[i];
      sX[(i + half * 8 + 16) * LDX + n0 + m] += c1[i];
    }
  }
  __syncthreads();

  rmsnorm(norm_w);

  // ---- Masked mean pool (NP patches x 64 dims) ----
  for (int o = t; o < NP * 64; o += 128) {
    const int pi = o >> 6, d = o & 63;
    int patch = pbase + pi;
    if (patch >= npatch) patch = npatch - 1;
    const int len = lengths[patch];
    float s = 0.f;
#pragma unroll
    for (int p = 0; p < NTOK; ++p)
      if (p < len) s += sXN[(pi * 8 + p) * LDX + d];
    sPool[pi * 64 + d] = s / (float)len;
  }
  __syncthreads();

  // ---- Final projection: out[NP][128] = pooled @ proj_w + proj_b ----
  for (int o = t; o < NP * DM; o += 128) {
    const int pi = o >> 7, c = o & 127;
    const int patch = pbase + pi;
    if (patch >= npatch) continue;
    const float* pr = sPool + pi * 64;
    float acc = proj_b[c];
#pragma unroll
    for (int d = 0; d < 64; ++d) acc += pr[d] * proj_w[d * DM + c];
    out[(size_t)patch * DM + c] = acc;
  }
}

extern "C" void kernel_launch(void* const* d_in, const int* in_sizes, int n_in,
                              void* d_out, int out_size, void* d_ws,
                              size_t ws_size, hipStream_t stream) {
  const int* tokens = (const int*)d_in[0];
  const int* lengths = (const int*)d_in[1];
  const float* emb = (const float*)d_in[2];
  const float* qkv_w = (const float*)d_in[3];
  const float* w_o = (const float*)d_in[4];
  const float* norm1_w = (const float*)d_in[5];
  const float* norm2_w = (const float*)d_in[6];
  const float* ffn_w1 = (const float*)d_in[7];
  const float* ffn_w2 = (const float*)d_in[8];
  const float* norm_w = (const float*)d_in[9];
  const float* proj_w = (const float*)d_in[10];
  const float* proj_b = (const float*)d_in[11];
  float* out = (float*)d_out;

  const int npatch = in_sizes[1];  // lengths: one per patch
  const int nblocks = (npatch + NP - 1) / NP;

  local_encoder_kernel<<<dim3(nblocks), dim3(128), 0, stream>>>(
      tokens, lengths, emb, qkv_w, w_o, norm1_w, norm2_w, ffn_w1, ffn_w2,
      norm_w, proj_w, proj_b, out, npatch);
}